// GraphormerModel_70789650972752
// MI455X (gfx1250) — compile-verified
//
#include <hip/hip_runtime.h>
#include <math.h>

// ---------------- problem constants (from reference) ----------------
#define NNODE   8192        // B*MAXN
#define NEDGE   131072      // B*MAXN*DEG
#define DIMC    256
#define DEC     128
#define NHEAD   8
#define DHC     32          // DIM/HEADS
#define BBAT    32
#define MAXNN   256
#define NLAYER  4
#define NGAUSS  32
#define MAXSPD_ 64
#define MAXDEG_ 32

typedef float v2f __attribute__((ext_vector_type(2)));
typedef float v8f __attribute__((ext_vector_type(8)));

// ======================================================================
// Generic f32 WMMA GEMM:  D = epilogue( alpha * A(gather?) @ (W or W^T) )
// epilogue: +bias[col], +addsrc[m,col], act(0 none,1 relu,2 gelu,3 sigmoid),
//           then optional (*gvec[col]*gmul + bvec[col])
// z-batching: base += (z/zdiv)*hi + (z%zdiv)*lo  for A, W, D/addsrc.
// One wave computes a 32x(NT*16) tile (MT=2 row tiles) with
// V_WMMA_F32_16X16X4_F32, software-pipelined by one k-step.
// ======================================================================
struct GemmCfg {
  const float* A; const float* W; const float* bias; const float* addsrc;
  const float* gvec; const float* bvec;
  float gmul, alpha;
  float* D; const int* gather;
  int M, N, K, lda, ldw, ldd, act, zdiv;
  long zA_hi, zA_lo, zW_hi, zW_lo, zD_hi, zD_lo;
};

template <int TRANSB, int NT>
__global__ __launch_bounds__(256) void gemm_f32_wmma(GemmCfg c) {
  const int wave = threadIdx.x >> 5;
  const int lane = threadIdx.x & 31;
  const int row0 = blockIdx.y * 256 + wave * 32;     // 2 row tiles per wave
  if (row0 >= c.M) return;                           // wave-uniform
  const int col0 = blockIdx.x * 64;
  const int z = blockIdx.z;
  const int zq = z / c.zdiv, zr = z % c.zdiv;
  const float* A = c.A + (long)zq * c.zA_hi + (long)zr * c.zA_lo;
  const float* W = c.W + (long)zq * c.zW_hi + (long)zr * c.zW_lo;
  float*       D = c.D + (long)zq * c.zD_hi + (long)zr * c.zD_lo;
  const float* S = c.addsrc ? (c.addsrc + (long)zq * c.zD_hi + (long)zr * c.zD_lo)
                            : (const float*)0;

  const int ml    = lane & 15;
  const int khalf = (lane >> 4) << 1;                // 0 (lanes 0-15), 2 (16-31)
  const int mrow0 = row0 + ml;
  const int mrow1 = row0 + 16 + ml;
  const int arow0 = c.gather ? c.gather[mrow0] : mrow0;
  const int arow1 = c.gather ? c.gather[mrow1] : mrow1;
  const float* ap0 = A + (long)arow0 * c.lda + khalf;
  const float* ap1 = A + (long)arow1 * c.lda + khalf;

  const long bstep = TRANSB ? 4 : (long)4 * c.ldw;
  const float* bp[NT];
#pragma unroll
  for (int nt = 0; nt < NT; nt++)
    bp[nt] = TRANSB ? (W + (long)(col0 + nt * 16 + ml) * c.ldw + khalf)
                    : (W + (long)khalf * c.ldw + col0 + nt * 16 + ml);

  v8f acc0[NT] = {};
  v8f acc1[NT] = {};

  // preload k = 0 fragments
  v2f a0, a1, b[NT];
  a0.x = ap0[0]; a0.y = ap0[1];
  a1.x = ap1[0]; a1.y = ap1[1];
#pragma unroll
  for (int nt = 0; nt < NT; nt++) {
    if (TRANSB) { b[nt].x = bp[nt][0]; b[nt].y = bp[nt][1]; }
    else        { b[nt].x = bp[nt][0]; b[nt].y = bp[nt][c.ldw]; }
  }

  for (int k = 4; k < c.K; k += 4) {
    ap0 += 4; ap1 += 4;
    v2f na0, na1, nb[NT];
    na0.x = ap0[0]; na0.y = ap0[1];
    na1.x = ap1[0]; na1.y = ap1[1];
#pragma unroll
    for (int nt = 0; nt < NT; nt++) {
      bp[nt] += bstep;
      if (TRANSB) { nb[nt].x = bp[nt][0]; nb[nt].y = bp[nt][1]; }
      else        { nb[nt].x = bp[nt][0]; nb[nt].y = bp[nt][c.ldw]; }
    }
#pragma unroll
    for (int nt = 0; nt < NT; nt++) {
      acc0[nt] = __builtin_amdgcn_wmma_f32_16x16x4_f32(
          false, a0, false, b[nt], (short)0, acc0[nt], false, false);
      acc1[nt] = __builtin_amdgcn_wmma_f32_16x16x4_f32(
          false, a1, false, b[nt], (short)0, acc1[nt], false, false);
    }
    a0 = na0; a1 = na1;
#pragma unroll
    for (int nt = 0; nt < NT; nt++) b[nt] = nb[nt];
  }
#pragma unroll
  for (int nt = 0; nt < NT; nt++) {
    acc0[nt] = __builtin_amdgcn_wmma_f32_16x16x4_f32(
        false, a0, false, b[nt], (short)0, acc0[nt], false, false);
    acc1[nt] = __builtin_amdgcn_wmma_f32_16x16x4_f32(
        false, a1, false, b[nt], (short)0, acc1[nt], false, false);
  }

  // -------- epilogue --------
  const int colL = lane & 15;
#pragma unroll
  for (int mt = 0; mt < 2; mt++) {
    const int mBase = row0 + mt * 16 + ((lane >> 4) << 3); // lanes16-31: M=r+8
#pragma unroll
    for (int nt = 0; nt < NT; nt++) {
      const int col = col0 + nt * 16 + colL;
      const float bia = c.bias ? c.bias[col] : 0.0f;
#pragma unroll
      for (int r = 0; r < 8; r++) {
        const int m = mBase + r;
        float v = (mt ? acc1[nt][r] : acc0[nt][r]) * c.alpha + bia;
        if (S) v += S[(long)m * c.ldd + col];
        if      (c.act == 1) v = fmaxf(v, 0.0f);
        else if (c.act == 2) v = 0.5f * v * (1.0f + erff(v * 0.70710678118654752f));
        else if (c.act == 3) v = 1.0f / (1.0f + expf(-v));
        if (c.gvec) v = v * c.gvec[col] * c.gmul + c.bvec[col];
        D[(long)m * c.ldd + col] = v;
      }
    }
  }
}

// ---------------- LayerNorm: one wave per row ----------------
__global__ __launch_bounds__(256) void ln_rows(const float* in, float* out,
    const float* g, const float* b, int len, int rows) {
  const int wave = threadIdx.x >> 5, lane = threadIdx.x & 31;
  const int row = blockIdx.x * 8 + wave;
  if (row >= rows) return;
  const float* p = in + (long)row * len;
  const int per = len >> 5;
  float s = 0.0f;
  for (int i = 0; i < per; i++) s += p[lane + i * 32];
  for (int off = 16; off; off >>= 1) s += __shfl_xor(s, off, 32);
  const float mean = s / (float)len;
  float vv = 0.0f;
  for (int i = 0; i < per; i++) { float d = p[lane + i * 32] - mean; vv += d * d; }
  for (int off = 16; off; off >>= 1) vv += __shfl_xor(vv, off, 32);
  const float rs = 1.0f / sqrtf(vv / (float)len + 1e-5f);
  float* o = out + (long)row * len;
  for (int i = 0; i < per; i++) {
    const int cix = lane + i * 32;
    o[cix] = (p[cix] - mean) * rs * g[cix] + b[cix];
  }
}

// ---------------- small elementwise kernels ----------------
__global__ void fill_f32(float* p, float v, long n) {
  long t = (long)blockIdx.x * blockDim.x + threadIdx.x; if (t < n) p[t] = v;
}
__global__ void fill_u32(unsigned* p, unsigned v, long n) {
  long t = (long)blockIdx.x * blockDim.x + threadIdx.x; if (t < n) p[t] = v;
}
__global__ void add2_k(const float* a, const float* b, float* o, long n) {
  long t = (long)blockIdx.x * blockDim.x + threadIdx.x; if (t < n) o[t] = a[t] + b[t];
}
__global__ void add3_k(const float* a, const float* b, const float* cc, float* o, long n) {
  long t = (long)blockIdx.x * blockDim.x + threadIdx.x; if (t < n) o[t] = a[t] + b[t] + cc[t];
}
__global__ void acc_k(float* p, const float* q, long n) {
  long t = (long)blockIdx.x * blockDim.x + threadIdx.x; if (t < n) p[t] += q[t];
}

// degree count + degree-embedding residual
__global__ void deg_count_k(const int* src, int* deg, int e) {
  int t = blockIdx.x * blockDim.x + threadIdx.x;
  if (t < e) atomicAdd(&deg[src[t]], 1);
}
__global__ void add_deg_emb_k(const float* X, const int* deg, const float* emb,
                              float* out, long n) {
  long t = (long)blockIdx.x * blockDim.x + threadIdx.x; if (t >= n) return;
  int m = (int)(t >> 8), ccol = (int)(t & 255);
  int d = deg[m]; if (d > MAXDEG_ - 1) d = MAXDEG_ - 1;
  out[t] = X[t] + emb[d * DIMC + ccol];
}

// spatial embedding lookup into D[:,128:256]
__global__ void spd_embed_k(const int* spd, const float* tab, float* D, long n) {
  long t = (long)blockIdx.x * blockDim.x + threadIdx.x; if (t >= n) return;
  int e = (int)(t >> 7), ccol = (int)(t & 127);
  int s = spd[e]; if (s > MAXSPD_ - 1) s = MAXSPD_ - 1; if (s < 0) s = 0;
  D[(long)e * 256 + 128 + ccol] = tab[s * DEC + ccol];
}

// ea = gate*ea_h + (1-gate)*spd_e   (pre-LN)
__global__ void fuse_combine_k(const float* D, const float* G, float* EA, long n) {
  long t = (long)blockIdx.x * blockDim.x + threadIdx.x; if (t >= n) return;
  int e = (int)(t >> 7), ccol = (int)(t & 127);
  float g = G[t];
  float eh = D[(long)e * 256 + ccol];
  float sp = D[(long)e * 256 + 128 + ccol];
  EA[t] = g * eh + (1.0f - g) * sp;
}

// edge attention scores: s[e,h] = scale * q[dst] . (k[src] + eproj[e])
__global__ void edge_scores_k(const float* qn, const float* kn, const float* ep,
                              const int* src, const int* dst, float scale,
                              float* sc, int eh) {
  int t = blockIdx.x * blockDim.x + threadIdx.x; if (t >= eh) return;
  int e = t >> 3, h = t & 7;
  const float* q = qn + (long)dst[e] * DIMC + h * DHC;
  const float* k = kn + (long)src[e] * DIMC + h * DHC;
  const float* p = ep + (long)e   * DIMC + h * DHC;
  float acc = 0.0f;
#pragma unroll
  for (int i = 0; i < DHC; i++) acc += q[i] * (k[i] + p[i]);
  sc[t] = acc * scale;
}

__device__ __forceinline__ unsigned f2key(float v) {
  unsigned iv = __float_as_uint(v);
  return (iv & 0x80000000u) ? ~iv : (iv | 0x80000000u);
}
__global__ void seg_max_k(const float* sc, const int* dst, unsigned* mk, int eh) {
  int t = blockIdx.x * blockDim.x + threadIdx.x; if (t >= eh) return;
  int e = t >> 3, h = t & 7;
  atomicMax(&mk[dst[e] * NHEAD + h], f2key(sc[t]));
}
__global__ void decode_max_k(const unsigned* mk, float* mf, int n) {
  int t = blockIdx.x * blockDim.x + threadIdx.x; if (t >= n) return;
  unsigned key = mk[t];
  if (key == 0u) { mf[t] = 0.0f; return; }
  unsigned iv = (key & 0x80000000u) ? (key & 0x7fffffffu) : ~key;
  float v = __uint_as_float(iv);
  mf[t] = isfinite(v) ? v : 0.0f;
}
__global__ void seg_exp_k(const float* sc, const float* mf, const int* dst,
                          float* al, float* sum, int eh) {
  int t = blockIdx.x * blockDim.x + threadIdx.x; if (t >= eh) return;
  int e = t >> 3, h = t & 7;
  float v = expf(sc[t] - mf[dst[e] * NHEAD + h]);
  al[t] = v;
  atomicAdd(&sum[dst[e] * NHEAD + h], v);
}
__global__ void seg_norm_k(float* al, const float* sum, const int* dst, int eh) {
  int t = blockIdx.x * blockDim.x + threadIdx.x; if (t >= eh) return;
  int e = t >> 3, h = t & 7;
  al[t] = al[t] / (sum[dst[e] * NHEAD + h] + 1e-16f);
}
// out[dst] += (v[src]+eproj[e]) * alpha[e,h]
__global__ void attn_accum_k(const float* vn, const float* ep, const float* al,
                             const int* src, const int* dst, float* out, long n) {
  long t = (long)blockIdx.x * blockDim.x + threadIdx.x; if (t >= n) return;
  int e = (int)(t >> 8), ccol = (int)(t & 255);
  int h = ccol >> 5;
  float v = (vn[(long)src[e] * DIMC + ccol] + ep[(long)e * DIMC + ccol]) * al[e * NHEAD + h];
  atomicAdd(&out[(long)dst[e] * DIMC + ccol], v);
}

// ea += sigmoid(gl) * delta  (raw D holds [delta | gl] with row stride 256)
__global__ void ea_update_k(const float* D, float* EA, long n) {
  long t = (long)blockIdx.x * blockDim.x + threadIdx.x; if (t >= n) return;
  int e = (int)(t >> 7), ccol = (int)(t & 127);
  float delta = D[(long)e * 256 + ccol];
  float gl    = D[(long)e * 256 + 128 + ccol];
  EA[t] += delta / (1.0f + expf(-gl));
}

// dense SPD bias accumulation
__global__ void dist_accum_k(const int* src, const int* dst, const int* spd,
                             int* dist, int e) {
  int t = blockIdx.x * blockDim.x + threadIdx.x; if (t >= e) return;
  int s = src[t], d = dst[t];
  int b = s >> 8;
  atomicAdd(&dist[((long)b << 16) + ((long)(s & 255) << 8) + (d & 255)], spd[t]);
}

// bias[b,h,i,j] = dist_emb[clip(dist)][h] + rbf(d3) @ rbf_W + rbf_b
__global__ void bias_build_k(const int* dist, const float* pos, const float* demb,
                             const float* rW, const float* rb, float* SCB, long n) {
  long t = (long)blockIdx.x * blockDim.x + threadIdx.x; if (t >= n) return;
  int b = (int)(t >> 16), i = (int)((t >> 8) & 255), j = (int)(t & 255);
  int dv = dist[t]; dv = dv < 0 ? 0 : (dv > MAXSPD_ - 1 ? MAXSPD_ - 1 : dv);
  const float* pi = pos + (long)(b * MAXNN + i) * 3;
  const float* pj = pos + (long)(b * MAXNN + j) * 3;
  float dx = pi[0] - pj[0], dy = pi[1] - pj[1], dz = pi[2] - pj[2];
  float d3 = sqrtf(dx * dx + dy * dy + dz * dz + 1e-12f);
  const float step = 20.0f / 31.0f;
  const float coeff = -0.5f / (step * step);
  float outv[NHEAD];
#pragma unroll
  for (int h = 0; h < NHEAD; h++) outv[h] = demb[dv * NHEAD + h] + rb[h];
  for (int g = 0; g < NGAUSS; g++) {
    float df = d3 - step * (float)g;
    float r = expf(coeff * df * df);
#pragma unroll
    for (int h = 0; h < NHEAD; h++) outv[h] += r * rW[g * NHEAD + h];
  }
#pragma unroll
  for (int h = 0; h < NHEAD; h++)
    SCB[(long)b * 524288 + (long)h * 65536 + (long)i * 256 + j] = outv[h];
}

// row softmax, one wave per 256-wide row (in place)
__global__ __launch_bounds__(256) void softmax_rows_k(float* p, long rows) {
  const int wave = threadIdx.x >> 5, lane = threadIdx.x & 31;
  long row = (long)blockIdx.x * 8 + wave;
  if (row >= rows) return;
  float* q = p + row * 256;
  float x[8];
#pragma unroll
  for (int i = 0; i < 8; i++) x[i] = q[lane + i * 32];
  float m = x[0];
#pragma unroll
  for (int i = 1; i < 8; i++) m = fmaxf(m, x[i]);
  for (int off = 16; off; off >>= 1) m = fmaxf(m, __shfl_xor(m, off, 32));
  float s = 0.0f;
#pragma unroll
  for (int i = 0; i < 8; i++) { x[i] = expf(x[i] - m); s += x[i]; }
  for (int off = 16; off; off >>= 1) s += __shfl_xor(s, off, 32);
  float inv = 1.0f / s;
#pragma unroll
  for (int i = 0; i < 8; i++) q[lane + i * 32] = x[i] * inv;
}

// ======================================================================
// host side
// ======================================================================
static void launch_gemm(hipStream_t st, const float* A, const float* W,
    const float* bias, const float* addsrc, const float* gvec, const float* bvec,
    float gmul, float alpha, float* D, const int* gather,
    int M, int N, int K, int lda, int ldw, int ldd, int transb, int act,
    int gz = 1, int zdiv = 1,
    long zAh = 0, long zAl = 0, long zWh = 0, long zWl = 0,
    long zDh = 0, long zDl = 0) {
  GemmCfg c{A, W, bias, addsrc, gvec, bvec, gmul, alpha, D, gather,
            M, N, K, lda, ldw, ldd, act, zdiv,
            zAh, zAl, zWh, zWl, zDh, zDl};
  dim3 g((N + 63) / 64, (M + 255) / 256, gz);
  const bool nt4 = (N % 64) == 0;
  if (transb) {
    if (nt4) hipLaunchKernelGGL((gemm_f32_wmma<1, 4>), g, dim3(256), 0, st, c);
    else     hipLaunchKernelGGL((gemm_f32_wmma<1, 2>), g, dim3(256), 0, st, c);
  } else {
    if (nt4) hipLaunchKernelGGL((gemm_f32_wmma<0, 4>), g, dim3(256), 0, st, c);
    else     hipLaunchKernelGGL((gemm_f32_wmma<0, 2>), g, dim3(256), 0, st, c);
  }
}
static inline int gsz(long n) { return (int)((n + 255) / 256); }

extern "C" void kernel_launch(void* const* d_in, const int* in_sizes, int n_in,
                              void* d_out, int out_size, void* d_ws, size_t ws_size,
                              hipStream_t stream) {
  (void)in_sizes; (void)n_in; (void)out_size; (void)ws_size;
  const float* IN_x   = (const float*)d_in[0];
  const float* IN_ea  = (const float*)d_in[1];
  const float* IN_pos = (const float*)d_in[2];
  int id = 3;
  const float* np_W = (const float*)d_in[id++]; const float* np_b = (const float*)d_in[id++];
  const float* np_g = (const float*)d_in[id++]; const float* np_be = (const float*)d_in[id++];
  const float* ep_W = (const float*)d_in[id++]; const float* ep_b = (const float*)d_in[id++];
  const float* ep_g = (const float*)d_in[id++]; const float* ep_be = (const float*)d_in[id++];
  const float* sp_emb = (const float*)d_in[id++];
  const float* fu_Wg = (const float*)d_in[id++]; const float* fu_bg = (const float*)d_in[id++];
  const float* fu_g  = (const float*)d_in[id++]; const float* fu_be = (const float*)d_in[id++];
  const float* L_[NLAYER][23];
  for (int l = 0; l < NLAYER; l++)
    for (int k = 0; k < 23; k++) L_[l][k] = (const float*)d_in[id++];
  const float* g_ln1g = (const float*)d_in[id++]; const float* g_ln1b = (const float*)d_in[id++];
  const float* g_ln2g = (const float*)d_in[id++]; const float* g_ln2b = (const float*)d_in[id++];
  const float* g_Wq = (const float*)d_in[id++];  const float* g_bq = (const float*)d_in[id++];
  const float* g_Wk = (const float*)d_in[id++];  const float* g_bk = (const float*)d_in[id++];
  const float* g_Wv = (const float*)d_in[id++];  const float* g_bv = (const float*)d_in[id++];
  const float* g_Wo = (const float*)d_in[id++];  const float* g_bo = (const float*)d_in[id++];
  const float* g_demb = (const float*)d_in[id++];
  const float* g_rW = (const float*)d_in[id++];  const float* g_rb = (const float*)d_in[id++];
  const float* g_W1 = (const float*)d_in[id++];  const float* g_b1 = (const float*)d_in[id++];
  const float* g_W2 = (const float*)d_in[id++];  const float* g_b2 = (const float*)d_in[id++];
  const int* spd = (const int*)d_in[id++];
  const int* eix = (const int*)d_in[id++];
  id++; // batch (unused)
  const int* src = eix;
  const int* dst = eix + NEDGE;

  // ---------------- workspace layout ----------------
  char* wp = (char*)d_ws;
  auto alloc = [&](size_t bytes) -> void* {
    void* p = (void*)wp; wp += (bytes + 255) & ~(size_t)255; return p;
  };
  const long NB = (long)NNODE * DIMC;           // 2,097,152 floats
  float* X    = (float*)alloc(NB * 4);
  float* COMB = (float*)alloc(NB * 4);
  float* TXIN = (float*)alloc(NB * 4);
  float* TH   = (float*)alloc(NB * 4);
  float* TH2  = (float*)alloc(NB * 4);
  float* TQ   = (float*)alloc(NB * 4);
  float* TK   = (float*)alloc(NB * 4);
  float* TV   = (float*)alloc(NB * 4);
  float* TS   = (float*)alloc(NB * 4);
  float* TO   = (float*)alloc(NB * 4);
  float* TXA  = (float*)alloc(NB * 4);
  float* TXN  = (float*)alloc(NB * 4);
  float* HG   = (float*)alloc(NB * 4);
  float* MID  = (float*)alloc((long)NNODE * 1024 * 4);
  float* EA   = (float*)alloc((long)NEDGE * DEC * 4);
  float* DBUF = (float*)alloc((long)NEDGE * 256 * 4);
  float* SC   = (float*)alloc((long)NEDGE * NHEAD * 4);
  float* AL   = (float*)alloc((long)NEDGE * NHEAD * 4);
  unsigned* MK = (unsigned*)alloc((long)NNODE * NHEAD * 4);
  float* MF   = (float*)alloc((long)NNODE * NHEAD * 4);
  float* SUM8 = (float*)alloc((long)NNODE * NHEAD * 4);
  int* DEG    = (int*)alloc((long)NNODE * 4);
  int* DIST   = (int*)alloc((long)BBAT * 65536 * 4);
  float* SCB  = (float*)alloc((long)BBAT * NHEAD * 65536 * 4);   // 67 MB
  float* GBUF = SCB;   // gate buffer aliases SCB (disjoint lifetimes)

  const float bninv = 0.99999500003749968f;     // 1/sqrt(1+1e-5)
  const float scale = 0.17677669529663687f;     // 1/sqrt(32)
  const long EH = (long)NEDGE * NHEAD;

  // ---------------- input projections ----------------
  // X = relu(x @ npW + npb) * (np_g*inv) + np_beta
  launch_gemm(stream, IN_x, np_W, np_b, nullptr, np_g, np_be, bninv, 1.0f,
              X, nullptr, NNODE, DIMC, 64, 64, DIMC, DIMC, 0, 1);
  // ea_h -> DBUF[:, :128]
  launch_gemm(stream, IN_ea, ep_W, ep_b, nullptr, ep_g, ep_be, bninv, 1.0f,
              DBUF, nullptr, NEDGE, DEC, 16, 16, DEC, 256, 0, 1);
  // spd embedding -> DBUF[:, 128:]
  hipLaunchKernelGGL(spd_embed_k, dim3(gsz((long)NEDGE * DEC)), dim3(256), 0, stream,
                     spd, sp_emb, DBUF, (long)NEDGE * DEC);
  // gate = sigmoid(DBUF @ fu_Wg + fu_bg)
  launch_gemm(stream, DBUF, fu_Wg, fu_bg, nullptr, nullptr, nullptr, 1.0f, 1.0f,
              GBUF, nullptr, NEDGE, DEC, 256, 256, DEC, DEC, 0, 3);
  hipLaunchKernelGGL(fuse_combine_k, dim3(gsz((long)NEDGE * DEC)), dim3(256), 0, stream,
                     DBUF, GBUF, EA, (long)NEDGE * DEC);
  hipLaunchKernelGGL(ln_rows, dim3(NEDGE / 8), dim3(256), 0, stream,
                     EA, EA, fu_g, fu_be, DEC, NEDGE);
  // degrees
  hipLaunchKernelGGL(fill_u32, dim3(gsz(NNODE)), dim3(256), 0, stream,
                     (unsigned*)DEG, 0u, (long)NNODE);
  hipLaunchKernelGGL(deg_count_k, dim3(gsz(NEDGE)), dim3(256), 0, stream, src, DEG, NEDGE);
  hipLaunchKernelGGL(fill_f32, dim3(gsz(NB)), dim3(256), 0, stream, COMB, 0.0f, NB);

  // ---------------- transformer-conv layers ----------------
  for (int l = 0; l < NLAYER; l++) {
    const float* ln_g = L_[l][0];  const float* ln_b = L_[l][1];
    const float* Wq = L_[l][2];    const float* bq = L_[l][3];
    const float* Wk = L_[l][4];    const float* bk = L_[l][5];
    const float* Wv = L_[l][6];    const float* bv = L_[l][7];
    const float* We = L_[l][8];    const float* be = L_[l][9];
    const float* Ws = L_[l][10];   const float* bs = L_[l][11];
    const float* W1 = L_[l][12];   const float* b1 = L_[l][13];
    const float* W2 = L_[l][14];   const float* b2 = L_[l][15];
    const float* demb = L_[l][16];
    const float* eu_ng = L_[l][17]; const float* eu_nb = L_[l][18];
    const float* eu_W = L_[l][19];  const float* eu_b = L_[l][20];
    const float* eu_eg = L_[l][21]; const float* eu_eb = L_[l][22];

    hipLaunchKernelGGL(add_deg_emb_k, dim3(gsz(NB)), dim3(256), 0, stream,
                       X, DEG, demb, TXIN, NB);
    hipLaunchKernelGGL(ln_rows, dim3(NNODE / 8), dim3(256), 0, stream,
                       TXIN, TH, ln_g, ln_b, DIMC, NNODE);
    launch_gemm(stream, TH, Wq, bq, nullptr, nullptr, nullptr, 1.0f, 1.0f,
                TQ, nullptr, NNODE, DIMC, DIMC, DIMC, DIMC, DIMC, 0, 0);
    launch_gemm(stream, TH, Wk, bk, nullptr, nullptr, nullptr, 1.0f, 1.0f,
                TK, nullptr, NNODE, DIMC, DIMC, DIMC, DIMC, DIMC, 0, 0);
    launch_gemm(stream, TH, Wv, bv, nullptr, nullptr, nullptr, 1.0f, 1.0f,
                TV, nullptr, NNODE, DIMC, DIMC, DIMC, DIMC, DIMC, 0, 0);
    launch_gemm(stream, TH, Ws, bs, nullptr, nullptr, nullptr, 1.0f, 1.0f,
                TS, nullptr, NNODE, DIMC, DIMC, DIMC, DIMC, DIMC, 0, 0);
    // eproj = ea @ We + be
    launch_gemm(stream, EA, We, be, nullptr, nullptr, nullptr, 1.0f, 1.0f,
                DBUF, nullptr, NEDGE, DIMC, DEC, DEC, DIMC, 256, 0, 0);
    // segment softmax attention
    hipLaunchKernelGGL(edge_scores_k, dim3(gsz(EH)), dim3(256), 0, stream,
                       TQ, TK, DBUF, src, dst, scale, SC, (int)EH);
    hipLaunchKernelGGL(fill_u32, dim3(gsz((long)NNODE * NHEAD)), dim3(256), 0, stream,
                       MK, 0u, (long)NNODE * NHEAD);
    hipLaunchKernelGGL(seg_max_k, dim3(gsz(EH)), dim3(256), 0, stream, SC, dst, MK, (int)EH);
    hipLaunchKernelGGL(decode_max_k, dim3(gsz((long)NNODE * NHEAD)), dim3(256), 0, stream,
                       MK, MF, NNODE * NHEAD);
    hipLaunchKernelGGL(fill_f32, dim3(gsz((long)NNODE * NHEAD)), dim3(256), 0, stream,
                       SUM8, 0.0f, (long)NNODE * NHEAD);
    hipLaunchKernelGGL(seg_exp_k, dim3(gsz(EH)), dim3(256), 0, stream,
                       SC, MF, dst, AL, SUM8, (int)EH);
    hipLaunchKernelGGL(seg_norm_k, dim3(gsz(EH)), dim3(256), 0, stream,
                       AL, SUM8, dst, (int)EH);
    hipLaunchKernelGGL(fill_f32, dim3(gsz(NB)), dim3(256), 0, stream, TO, 0.0f, NB);
    hipLaunchKernelGGL(attn_accum_k, dim3(gsz((long)NEDGE * DIMC)), dim3(256), 0, stream,
                       TV, DBUF, AL, src, dst, TO, (long)NEDGE * DIMC);
    // x_attn = x_in + out + (h@Ws+bs)
    hipLaunchKernelGGL(add3_k, dim3(gsz(NB)), dim3(256), 0, stream, TXIN, TO, TS, TXA, NB);
    // FFN
    hipLaunchKernelGGL(ln_rows, dim3(NNODE / 8), dim3(256), 0, stream,
                       TXA, TH2, ln_g, ln_b, DIMC, NNODE);
    launch_gemm(stream, TH2, W1, b1, nullptr, nullptr, nullptr, 1.0f, 1.0f,
                MID, nullptr, NNODE, 1024, DIMC, DIMC, 1024, 1024, 0, 2);
    launch_gemm(stream, MID, W2, b2, TXA, nullptr, nullptr, 1.0f, 1.0f,
                X, nullptr, NNODE, DIMC, 1024, 1024, DIMC, DIMC, 0, 0);
    if (l >= 1)
      hipLaunchKernelGGL(acc_k, dim3(gsz(NB)), dim3(256), 0, stream, COMB, X, NB);
    // edge update: raw = relu(xn[src]@Wa + xn[dst]@Wb + ea@Wc + eu_b)
    hipLaunchKernelGGL(ln_rows, dim3(NNODE / 8), dim3(256), 0, stream,
                       X, TXN, eu_ng, eu_nb, DIMC, NNODE);
    launch_gemm(stream, TXN, eu_W, nullptr, nullptr, nullptr, nullptr, 1.0f, 1.0f,
                DBUF, src, NEDGE, 256, DIMC, DIMC, 256, 256, 0, 0);
    launch_gemm(stream, TXN, eu_W + 256 * 256, nullptr, DBUF, nullptr, nullptr, 1.0f, 1.0f,
                DBUF, dst, NEDGE, 256, DIMC, DIMC, 256, 256, 0, 0);
    launch_gemm(stream, EA, eu_W + 512 * 256, eu_b, DBUF, nullptr, nullptr, 1.0f, 1.0f,
                DBUF, nullptr, NEDGE, 256, DEC, DEC, 256, 256, 0, 1);
    hipLaunchKernelGGL(ea_update_k, dim3(gsz((long)NEDGE * DEC)), dim3(256), 0, stream,
                       DBUF, EA, (long)NEDGE * DEC);
    hipLaunchKernelGGL(ln_rows, dim3(NEDGE / 8), dim3(256), 0, stream,
                       EA, EA, eu_eg, eu_eb, DEC, NEDGE);
  }

  // ---------------- global attention layer ----------------
  hipLaunchKernelGGL(fill_u32, dim3(gsz((long)BBAT * 65536)), dim3(256), 0, stream,
                     (unsigned*)DIST, 0u, (long)BBAT * 65536);
  hipLaunchKernelGGL(dist_accum_k, dim3(gsz(NEDGE)), dim3(256), 0, stream,
                     src, dst, spd, DIST, NEDGE);
  hipLaunchKernelGGL(bias_build_k, dim3(gsz((long)BBAT * 65536)), dim3(256), 0, stream,
                     DIST, IN_pos, g_demb, g_rW, g_rb, SCB, (long)BBAT * 65536);
  hipLaunchKernelGGL(ln_rows, dim3(NNODE / 8), dim3(256), 0, stream,
                     COMB, TH, g_ln1g, g_ln1b, DIMC, NNODE);
  launch_gemm(stream, TH, g_Wq, g_bq, nullptr, nullptr, nullptr, 1.0f, 1.0f,
              TQ, nullptr, NNODE, DIMC, DIMC, DIMC, DIMC, DIMC, 0, 0);
  launch_gemm(stream, TH, g_Wk, g_bk, nullptr, nullptr, nullptr, 1.0f, 1.0f,
              TK, nullptr, NNODE, DIMC, DIMC, DIMC, DIMC, DIMC, 0, 0);
  launch_gemm(stream, TH, g_Wv, g_bv, nullptr, nullptr, nullptr, 1.0f, 1.0f,
              TV, nullptr, NNODE, DIMC, DIMC, DIMC, DIMC, DIMC, 0, 0);
  // SCB[b,h] += scale * Q_bh @ K_bh^T   (batched over z = b*8+h)
  launch_gemm(stream, TQ, TK, nullptr, SCB, nullptr, nullptr, 1.0f, scale,
              SCB, nullptr, MAXNN, MAXNN, DHC, DIMC, DIMC, MAXNN, 1, 0,
              BBAT * NHEAD, NHEAD,
              65536, 32, 65536, 32, 524288, 65536);
  hipLaunchKernelGGL(softmax_rows_k, dim3((BBAT * NHEAD * MAXNN) / 8), dim3(256), 0, stream,
                     SCB, (long)BBAT * NHEAD * MAXNN);
  // TO[b, :, h*32:(h+1)*32] = P_bh @ V_bh  (batched)
  launch_gemm(stream, SCB, TV, nullptr, nullptr, nullptr, nullptr, 1.0f, 1.0f,
              TO, nullptr, MAXNN, DHC, MAXNN, MAXNN, DIMC, DIMC, 0, 0,
              BBAT * NHEAD, NHEAD,
              524288, 65536, 65536, 32, 65536, 32);
  // H = TH + TO @ Wo + bo
  launch_gemm(stream, TO, g_Wo, g_bo, TH, nullptr, nullptr, 1.0f, 1.0f,
              HG, nullptr, NNODE, DIMC, DIMC, DIMC, DIMC, DIMC, 0, 0);
  // H += gelu(ln(H) @ W1 + b1) @ W2 + b2
  hipLaunchKernelGGL(ln_rows, dim3(NNODE / 8), dim3(256), 0, stream,
                     HG, TH2, g_ln2g, g_ln2b, DIMC, NNODE);
  launch_gemm(stream, TH2, g_W1, g_b1, nullptr, nullptr, nullptr, 1.0f, 1.0f,
              MID, nullptr, NNODE, 1024, DIMC, DIMC, 1024, 1024, 0, 2);
  launch_gemm(stream, MID, g_W2, g_b2, HG, nullptr, nullptr, 1.0f, 1.0f,
              HG, nullptr, NNODE, DIMC, 1024, 1024, DIMC, DIMC, 0, 0);
  // out = comb + H
  hipLaunchKernelGGL(add2_k, dim3(gsz(NB)), dim3(256), 0, stream,
                     COMB, HG, (float*)d_out, NB);
}